// VoxelModule_54365696033236
// MI455X (gfx1250) — compile-verified
//
#include <hip/hip_runtime.h>

// Problem constants (fixed by the reference setup)
#define VOX        40
#define V3         (VOX * VOX * VOX)      // 64000 voxels per batch
#define NBATCH     32
#define NPTS       500000
#define MM_BLOCKS  128                    // min/max partial blocks per batch
#define MM_THREADS 256
#define HIST_BLKS  8                      // histogram blocks per batch (N % 8 == 0)
#define HIST_THREADS 1024
#define TILE_PTS   1024                   // points staged per async tile
#define TILE_FLOATS (TILE_PTS * 3)        // 3072 floats = 12 KB per buffer

// Pointer types for the async global->LDS DMA builtin (prototype revealed by
// the round-2 diagnostic: AS1 v4i* source; AS3 v4i* LDS destination).
typedef int v4i __attribute__((vector_size(16)));
typedef __attribute__((address_space(1))) v4i* gbl_v4i_p;
typedef __attribute__((address_space(3))) v4i* lds_v4i_p;

// ---------------------------------------------------------------------------
// K1: zero the int32 counts workspace (harness does not re-poison between runs)
// ---------------------------------------------------------------------------
__global__ void k_zero_counts(int* __restrict__ counts, int n) {
    int i = blockIdx.x * blockDim.x + threadIdx.x;
    int stride = gridDim.x * blockDim.x;
    for (; i < n; i += stride) counts[i] = 0;
}

// ---------------------------------------------------------------------------
// K2: per-batch partial min/max.  grid = (MM_BLOCKS, NBATCH), 256 threads.
// fmin/fmax are exact & order-independent -> deterministic.
// ---------------------------------------------------------------------------
__global__ void k_partial_minmax(const float* __restrict__ pc,
                                 float* __restrict__ pmin,
                                 float* __restrict__ pmax) {
    const int b   = blockIdx.y;
    const int tid = threadIdx.x;
    const int t0  = blockIdx.x * blockDim.x + tid;
    const int stride = gridDim.x * blockDim.x;
    const float* base = pc + (size_t)b * NPTS * 3;

    float mnx =  __builtin_inff(), mny =  __builtin_inff(), mnz =  __builtin_inff();
    float mxx = -__builtin_inff(), mxy = -__builtin_inff(), mxz = -__builtin_inff();

    for (int p = t0; p < NPTS; p += stride) {
        const float* q = base + 3 * (size_t)p;
        __builtin_prefetch(base + 3 * (size_t)(p + 4 * stride), 0, 0);
        float x = q[0], y = q[1], z = q[2];
        mnx = fminf(mnx, x); mxx = fmaxf(mxx, x);
        mny = fminf(mny, y); mxy = fmaxf(mxy, y);
        mnz = fminf(mnz, z); mxz = fmaxf(mxz, z);
    }

    __shared__ float smn[3][MM_THREADS];
    __shared__ float smx[3][MM_THREADS];
    smn[0][tid] = mnx; smn[1][tid] = mny; smn[2][tid] = mnz;
    smx[0][tid] = mxx; smx[1][tid] = mxy; smx[2][tid] = mxz;
    __syncthreads();
    for (int off = MM_THREADS / 2; off > 0; off >>= 1) {
        if (tid < off) {
            smn[0][tid] = fminf(smn[0][tid], smn[0][tid + off]);
            smn[1][tid] = fminf(smn[1][tid], smn[1][tid + off]);
            smn[2][tid] = fminf(smn[2][tid], smn[2][tid + off]);
            smx[0][tid] = fmaxf(smx[0][tid], smx[0][tid + off]);
            smx[1][tid] = fmaxf(smx[1][tid], smx[1][tid + off]);
            smx[2][tid] = fmaxf(smx[2][tid], smx[2][tid + off]);
        }
        __syncthreads();
    }
    if (tid == 0) {
        size_t o = ((size_t)b * MM_BLOCKS + blockIdx.x) * 3;
        pmin[o + 0] = smn[0][0]; pmin[o + 1] = smn[1][0]; pmin[o + 2] = smn[2][0];
        pmax[o + 0] = smx[0][0]; pmax[o + 1] = smx[1][0]; pmax[o + 2] = smx[2][0];
    }
}

// ---------------------------------------------------------------------------
// K3: final min/max reduce + per-batch params {min.xyz, range.xyz}.
// ---------------------------------------------------------------------------
__global__ void k_final_minmax(const float* __restrict__ pmin,
                               const float* __restrict__ pmax,
                               float* __restrict__ params) {
    const int b   = blockIdx.x;
    const int tid = threadIdx.x;
    size_t o = ((size_t)b * MM_BLOCKS + tid) * 3;
    float mnx = pmin[o + 0], mny = pmin[o + 1], mnz = pmin[o + 2];
    float mxx = pmax[o + 0], mxy = pmax[o + 1], mxz = pmax[o + 2];

    __shared__ float smn[3][MM_BLOCKS];
    __shared__ float smx[3][MM_BLOCKS];
    smn[0][tid] = mnx; smn[1][tid] = mny; smn[2][tid] = mnz;
    smx[0][tid] = mxx; smx[1][tid] = mxy; smx[2][tid] = mxz;
    __syncthreads();
    for (int off = MM_BLOCKS / 2; off > 0; off >>= 1) {
        if (tid < off) {
            smn[0][tid] = fminf(smn[0][tid], smn[0][tid + off]);
            smn[1][tid] = fminf(smn[1][tid], smn[1][tid + off]);
            smn[2][tid] = fminf(smn[2][tid], smn[2][tid + off]);
            smx[0][tid] = fmaxf(smx[0][tid], smx[0][tid + off]);
            smx[1][tid] = fmaxf(smx[1][tid], smx[1][tid + off]);
            smx[2][tid] = fmaxf(smx[2][tid], smx[2][tid + off]);
        }
        __syncthreads();
    }
    if (tid == 0) {
        params[b * 8 + 0] = smn[0][0];
        params[b * 8 + 1] = smn[1][0];
        params[b * 8 + 2] = smn[2][0];
        params[b * 8 + 3] = smx[0][0] - smn[0][0];
        params[b * 8 + 4] = smx[1][0] - smn[1][0];
        params[b * 8 + 5] = smx[2][0] - smn[2][0];
    }
}

// ---------------------------------------------------------------------------
// Async-DMA one staging tile: dense b128 transfers global -> LDS (ASYNCcnt).
// ---------------------------------------------------------------------------
__device__ __forceinline__ void issue_tile(const float* __restrict__ gsrc,
                                           float* stage_buf,
                                           int fbase, int nfl, int tid) {
    if (tid * 4 < nfl) {
        __builtin_amdgcn_global_load_async_to_lds_b128(
            (gbl_v4i_p)(gsrc + fbase + tid * 4),
            (lds_v4i_p)(stage_buf + tid * 4),
            /*offset=*/0, /*cpol=*/0);
    }
}

// ---------------------------------------------------------------------------
// K4: fused normalize + flat-index + 64000-bin LDS histogram, with the point
// stream staged through LDS via double-buffered async DMA.
// grid = (HIST_BLKS, NBATCH), 1024 threads.
// LDS: 256,000 B hist + 24,576 B stage = 280,576 B  (<= 320 KB / WGP).
// ---------------------------------------------------------------------------
__global__ __launch_bounds__(HIST_THREADS, 1)
void k_voxelize_hist(const float* __restrict__ pc,
                     const float* __restrict__ params,
                     float* __restrict__ out,
                     int* __restrict__ counts) {
    __shared__ int hist[V3];                                       // 256,000 B
    __shared__ __attribute__((aligned(16))) float stage[2][TILE_FLOATS];

    const int b   = blockIdx.y;
    const int h   = blockIdx.x;
    const int tid = threadIdx.x;

    for (int i = tid; i < V3; i += blockDim.x) hist[i] = 0;

    const float mnx = params[b * 8 + 0];
    const float mny = params[b * 8 + 1];
    const float mnz = params[b * 8 + 2];
    const float rgx = params[b * 8 + 3];
    const float rgy = params[b * 8 + 4];
    const float rgz = params[b * 8 + 5];

    const int   chunk = NPTS / HIST_BLKS;          // 62500 points
    const int   p0    = h * chunk;
    const float* gsrc    = pc  + ((size_t)b * NPTS + p0) * 3;      // 16B-aligned
    float*      flat_out = out + ((size_t)b * NPTS + p0);
    float*      norm_out = out + (size_t)NBATCH * NPTS
                               + (size_t)NBATCH * V3
                               + ((size_t)b * NPTS + p0) * 3;

    const int nfl_total = chunk * 3;                               // 187500
    const int ntiles    = (chunk + TILE_PTS - 1) / TILE_PTS;       // 62

    // Prologue: start DMA of tile 0 into buffer 0.
    {
        int nfl = nfl_total < TILE_FLOATS ? nfl_total : TILE_FLOATS;
        issue_tile(gsrc, stage[0], 0, nfl, tid);
    }

    for (int k = 0; k < ntiles; ++k) {
        const int buf = k & 1;
        if (k + 1 < ntiles) {
            // Start DMA of tile k+1 into the other buffer (its last reader
            // finished at the barrier ending iteration k-1).
            int fbase = (k + 1) * TILE_FLOATS;
            int nfl = nfl_total - fbase;
            if (nfl > TILE_FLOATS) nfl = TILE_FLOATS;
            issue_tile(gsrc, stage[(k + 1) & 1], fbase, nfl, tid);
            // In-order completion: <=1 outstanding ==> tile k is resident.
            asm volatile("s_wait_asynccnt 0x1" ::: "memory");
        } else {
            asm volatile("s_wait_asynccnt 0x0" ::: "memory");
        }
        __syncthreads();   // tile k visible to all waves (also covers hist=0)

        const int pbase = k * TILE_PTS;
        int np = chunk - pbase;
        if (np > TILE_PTS) np = TILE_PTS;
        if (tid < np) {
            const float* sp = &stage[buf][tid * 3];
            float x = sp[0], y = sp[1], z = sp[2];

            float nx = (x - mnx) / rgx * (float)(VOX - 1);
            float ny = (y - mny) / rgy * (float)(VOX - 1);
            float nz = (z - mnz) / rgz * (float)(VOX - 1);

            int ix = (int)nx; ix = ix < 0 ? 0 : (ix > VOX - 1 ? VOX - 1 : ix);
            int iy = (int)ny; iy = iy < 0 ? 0 : (iy > VOX - 1 ? VOX - 1 : iy);
            int iz = (int)nz; iz = iz < 0 ? 0 : (iz > VOX - 1 ? VOX - 1 : iz);
            int flat = ix * (VOX * VOX) + iy * VOX + iz;

            atomicAdd(&hist[flat], 1);                      // ds_add_u32

            int p = pbase + tid;
            __builtin_nontemporal_store((float)flat, flat_out + p);
            __builtin_nontemporal_store(nx, norm_out + 3 * (size_t)p + 0);
            __builtin_nontemporal_store(ny, norm_out + 3 * (size_t)p + 1);
            __builtin_nontemporal_store(nz, norm_out + 3 * (size_t)p + 2);
        }
        __syncthreads();   // all reads of stage[buf] done before reuse at k+2
    }

    int* cb = counts + (size_t)b * V3;
    for (int i = tid; i < V3; i += blockDim.x) {
        int c = hist[i];
        if (c) atomicAdd(&cb[i], c);                        // global_atomic_add_u32
    }
}

// ---------------------------------------------------------------------------
// K5: convert int32 counts workspace -> float output region.
// ---------------------------------------------------------------------------
__global__ void k_counts_to_float(const int* __restrict__ counts,
                                  float* __restrict__ out) {
    int i = blockIdx.x * blockDim.x + threadIdx.x;
    int stride = gridDim.x * blockDim.x;
    const int n = NBATCH * V3;
    float* dst = out + (size_t)NBATCH * NPTS;
    for (; i < n; i += stride)
        __builtin_nontemporal_store((float)counts[i], dst + i);
}

// ---------------------------------------------------------------------------
extern "C" void kernel_launch(void* const* d_in, const int* in_sizes, int n_in,
                              void* d_out, int out_size, void* d_ws, size_t ws_size,
                              hipStream_t stream) {
    (void)in_sizes; (void)n_in; (void)out_size; (void)ws_size;
    const float* pc  = (const float*)d_in[0];
    float*       out = (float*)d_out;

    // Workspace layout
    int*   counts = (int*)d_ws;                                    // 8.19 MB
    float* pmin   = (float*)((char*)d_ws + sizeof(int) * (size_t)NBATCH * V3);
    float* pmax   = pmin + (size_t)NBATCH * MM_BLOCKS * 3;
    float* params = pmax + (size_t)NBATCH * MM_BLOCKS * 3;

    k_zero_counts<<<256, 256, 0, stream>>>(counts, NBATCH * V3);

    dim3 gmm(MM_BLOCKS, NBATCH);
    k_partial_minmax<<<gmm, MM_THREADS, 0, stream>>>(pc, pmin, pmax);

    k_final_minmax<<<NBATCH, MM_BLOCKS, 0, stream>>>(pmin, pmax, params);

    dim3 gvh(HIST_BLKS, NBATCH);
    k_voxelize_hist<<<gvh, HIST_THREADS, 0, stream>>>(pc, params, out, counts);

    k_counts_to_float<<<(NBATCH * V3 + 255) / 256, 256, 0, stream>>>(counts, out);
}